// SingleBatch_CNAPSProtoNetSimilarityModule_88622355185858
// MI455X (gfx1250) — compile-verified
//
#include <hip/hip_runtime.h>
#include <stdint.h>

// ---------------------------------------------------------------------------
// CNAPS Mahalanobis head for MI455X (gfx1250, wave32, WMMA f32 16x16x4).
//   support S=2048, query Q=8192, D=2048, C=2.
//   All GEMM-shaped work routed through V_WMMA_F32_16X16X4_F32.
//   Matrix inverse via Newton-Schulz (SPD, lambda_min >= 1 due to +I reg).
// ---------------------------------------------------------------------------

#define DD   2048
#define NSUP 2048
#define NQRY 8192
#define NTOT 10240
#define NCLS 2
#define NS_ITERS 12

typedef __attribute__((ext_vector_type(2))) float v2f;
typedef __attribute__((ext_vector_type(8))) float v8f;

// workspace byte offsets (all 256B+ aligned); total ~106 MB
#define OFF_SIDX  (0u)
#define OFF_QIDX  (64u * 1024u)
#define OFF_SCAL  (128u * 1024u)      // f[0..1]=n_k ; u[4..5]=||sigma||_1 bits
#define OFF_MU    (192u * 1024u)      // mu[2][D]
#define OFF_MEAN  (256u * 1024u)      // mean_all[D]
#define OFF_COVT  (1u  << 20)
#define OFF_SIG0  (18u << 20)
#define OFF_SIG1  (36u << 20)
#define OFF_XA    (54u << 20)
#define OFF_XB    (72u << 20)
#define OFF_TMP   (90u << 20)

__device__ __forceinline__ v8f wmma4(v2f a, v2f b, v8f c) {
  // D = A(16x4,f32) * B(4x16,f32) + C(16x16,f32)
  return __builtin_amdgcn_wmma_f32_16x16x4_f32(false, a, false, b, (short)0, c,
                                               false, false);
}

// ---------------------------------------------------------------------------
__global__ void k_init(unsigned* normbits) {
  if (threadIdx.x < 2) normbits[threadIdx.x] = 0u;
}

// Stream-compaction of support/query indices + class counts (single block).
__global__ void k_indices(const int* __restrict__ labels,
                          const unsigned char* __restrict__ isq,
                          int* __restrict__ sidx, int* __restrict__ qidx,
                          float* __restrict__ scal) {
  __shared__ unsigned wS[8], wQ[8];
  __shared__ unsigned baseS, baseQ;
  __shared__ int c0s, c1s;
  const int t = threadIdx.x, lane = t & 31, wv = t >> 5;
  if (t == 0) { baseS = 0u; baseQ = 0u; c0s = 0; c1s = 0; }
  __syncthreads();
  for (int chunk = 0; chunk < NTOT; chunk += 256) {
    const int i = chunk + t;
    const int fS = (isq[i] == 0);
    const int fQ = (isq[i] != 0);
    const unsigned long long bS = __ballot(fS);
    const unsigned long long bQ = __ballot(fQ);
    const unsigned long long lt =
        (lane == 0) ? 0ull : ((~0ull) >> (64 - lane));
    const unsigned offS = (unsigned)__popcll(bS & lt);
    const unsigned offQ = (unsigned)__popcll(bQ & lt);
    if (lane == 0) {
      wS[wv] = (unsigned)__popcll(bS);
      wQ[wv] = (unsigned)__popcll(bQ);
    }
    __syncthreads();
    if (t == 0) {
      unsigned rs = baseS, rq = baseQ;
      for (int w = 0; w < 8; ++w) {
        unsigned a = wS[w]; wS[w] = rs; rs += a;
        unsigned b = wQ[w]; wQ[w] = rq; rq += b;
      }
      baseS = rs; baseQ = rq;
    }
    __syncthreads();
    if (fS) sidx[wS[wv] + offS] = i;
    if (fQ) qidx[wQ[wv] + offQ] = i;
    __syncthreads();
  }
  for (int s = t; s < NSUP; s += 256) {
    const int lab = labels[sidx[s]];
    if (lab == 0) atomicAdd(&c0s, 1); else atomicAdd(&c1s, 1);
  }
  __syncthreads();
  if (t == 0) { scal[0] = (float)c0s; scal[1] = (float)c1s; }
}

// Per-column class means + global mean over support rows.
__global__ __launch_bounds__(256)
void k_colstats(const float* __restrict__ G, const int* __restrict__ labels,
                const int* __restrict__ sidx, const float* __restrict__ scal,
                float* __restrict__ mu, float* __restrict__ meanall) {
  __shared__ int ss[NSUP];
  __shared__ int ls[NSUP];
  const int t = threadIdx.x;
  for (int s = t; s < NSUP; s += 256) {
    const int r = sidx[s];
    ss[s] = r;
    ls[s] = labels[r];
  }
  __syncthreads();
  const int d = blockIdx.x * 256 + t;
  float s0 = 0.f, s1 = 0.f, sa = 0.f;
  for (int s = 0; s < NSUP; ++s) {
    const float v = G[(size_t)ss[s] * DD + d];
    sa += v;
    if (ls[s] == 0) s0 += v; else s1 += v;
  }
  mu[d]      = s0 / scal[0];
  mu[DD + d] = s1 / scal[1];
  meanall[d] = sa / (float)NSUP;
}

// ---------------------------------------------------------------------------
// Covariance GEMM: out[m,n] = sum_s w(s)*(G[s,m]-muA[m])*(G[s,n]-muB[n])
// over gathered support rows; w(s) = (labels==cls) or 1 if cls<0.
// Block = 128x128 tile of out; 8 waves, each wave 32x64 (2x4 WMMA tiles).
__global__ __launch_bounds__(256)
void k_gemm_cov(const float* __restrict__ G, const int* __restrict__ sidx,
                const int* __restrict__ labels, const float* __restrict__ muA,
                const float* __restrict__ muB, int cls,
                float* __restrict__ out) {
  __shared__ float As[128 * 18];   // A^T tile: [m][k], pitch 18
  __shared__ float Bs[16 * 136];   // B tile: [k][n], pitch 136
  __shared__ float muAs[128], muBs[128];
  __shared__ int   rows[16];
  __shared__ float wrow[16];
  const int t = threadIdx.x, lane = t & 31, wv = t >> 5;
  const int wm = wv >> 1, wn = wv & 1;
  const int bm = blockIdx.x * 128, bn = blockIdx.y * 128;
  const int hi = lane >> 4, ln = lane & 15;
  if (t < 128) muAs[t] = muA[bm + t];
  else         muBs[t - 128] = muB[bn + (t - 128)];
  v8f acc[2][4];
#pragma unroll
  for (int i = 0; i < 2; ++i)
#pragma unroll
    for (int j = 0; j < 4; ++j) acc[i][j] = {};
  __syncthreads();
  for (int kk = 0; kk < NSUP; kk += 16) {
    if (t < 16) {
      const int r = sidx[kk + t];
      rows[t] = r;
      wrow[t] = (cls < 0 || labels[r] == cls) ? 1.f : 0.f;
    }
    __syncthreads();
#pragma unroll
    for (int p = 0; p < 8; ++p) {          // A: gather + center + mask
      const int k = p * 2 + (t >> 7);
      const int m = t & 127;
      const float v = G[(size_t)rows[k] * DD + bm + m] - muAs[m];
      As[m * 18 + k] = wrow[k] * v;
    }
#pragma unroll
    for (int p = 0; p < 8; ++p) {          // B: gather + center
      const int k = p * 2 + (t >> 7);
      const int n = t & 127;
      Bs[k * 136 + n] = G[(size_t)rows[k] * DD + bn + n] - muBs[n];
    }
    __syncthreads();
#pragma unroll
    for (int k0 = 0; k0 < 16; k0 += 4) {
      const v2f a0 = *(const v2f*)&As[(wm * 32 + ln) * 18 + k0 + 2 * hi];
      const v2f a1 = *(const v2f*)&As[(wm * 32 + 16 + ln) * 18 + k0 + 2 * hi];
#pragma unroll
      for (int nt = 0; nt < 4; ++nt) {
        const int n = wn * 64 + nt * 16 + ln;
        v2f b;
        b.x = Bs[(k0 + 2 * hi) * 136 + n];
        b.y = Bs[(k0 + 2 * hi + 1) * 136 + n];
        acc[0][nt] = wmma4(a0, b, acc[0][nt]);
        acc[1][nt] = wmma4(a1, b, acc[1][nt]);
      }
    }
    __syncthreads();
  }
#pragma unroll
  for (int mt = 0; mt < 2; ++mt)
#pragma unroll
    for (int nt = 0; nt < 4; ++nt)
#pragma unroll
      for (int v = 0; v < 8; ++v) {
        const size_t r = (size_t)(bm + wm * 32 + mt * 16 + hi * 8 + v);
        const size_t c = (size_t)(bn + wn * 64 + nt * 16 + ln);
        out[r * DD + c] = acc[mt][nt][v];
      }
}

// Square GEMM: out = alpha*(A@B) + beta*Cin   (A,B,Cin,out row-major DxD)
__global__ __launch_bounds__(256)
void k_gemm_sq(const float* __restrict__ A, const float* __restrict__ B,
               const float* __restrict__ Cin, float alpha, float beta,
               float* __restrict__ out) {
  __shared__ float As[128 * 18];
  __shared__ float Bs[16 * 136];
  const int t = threadIdx.x, lane = t & 31, wv = t >> 5;
  const int wm = wv >> 1, wn = wv & 1;
  const int bm = blockIdx.x * 128, bn = blockIdx.y * 128;
  const int hi = lane >> 4, ln = lane & 15;
  v8f acc[2][4];
#pragma unroll
  for (int i = 0; i < 2; ++i)
#pragma unroll
    for (int j = 0; j < 4; ++j) acc[i][j] = {};
  for (int kk = 0; kk < DD; kk += 16) {
#pragma unroll
    for (int p = 0; p < 2; ++p) {          // A: 128x16, float4 loads
      const int slot = p * 256 + t;
      const int r = slot >> 2, kg = (slot & 3) << 2;
      const float4 av = *(const float4*)&A[(size_t)(bm + r) * DD + kk + kg];
      As[r * 18 + kg + 0] = av.x;
      As[r * 18 + kg + 1] = av.y;
      As[r * 18 + kg + 2] = av.z;
      As[r * 18 + kg + 3] = av.w;
    }
#pragma unroll
    for (int p = 0; p < 8; ++p) {          // B: 16x128 row-coalesced
      const int k = p * 2 + (t >> 7);
      const int n = t & 127;
      Bs[k * 136 + n] = B[(size_t)(kk + k) * DD + bn + n];
    }
    __syncthreads();
#pragma unroll
    for (int k0 = 0; k0 < 16; k0 += 4) {
      const v2f a0 = *(const v2f*)&As[(wm * 32 + ln) * 18 + k0 + 2 * hi];
      const v2f a1 = *(const v2f*)&As[(wm * 32 + 16 + ln) * 18 + k0 + 2 * hi];
#pragma unroll
      for (int nt = 0; nt < 4; ++nt) {
        const int n = wn * 64 + nt * 16 + ln;
        v2f b;
        b.x = Bs[(k0 + 2 * hi) * 136 + n];
        b.y = Bs[(k0 + 2 * hi + 1) * 136 + n];
        acc[0][nt] = wmma4(a0, b, acc[0][nt]);
        acc[1][nt] = wmma4(a1, b, acc[1][nt]);
      }
    }
    __syncthreads();
  }
#pragma unroll
  for (int mt = 0; mt < 2; ++mt)
#pragma unroll
    for (int nt = 0; nt < 4; ++nt)
#pragma unroll
      for (int v = 0; v < 8; ++v) {
        const size_t idx =
            (size_t)(bm + wm * 32 + mt * 16 + hi * 8 + v) * DD +
            (size_t)(bn + wn * 64 + nt * 16 + ln);
        float r = alpha * acc[mt][nt][v];
        if (beta != 0.f) r += beta * Cin[idx];
        out[idx] = r;
      }
}

// sigma_c = lam*cov_k/(n-1) + (1-lam)*cov_t/(S-1) + I   (in place on raw sums)
__global__ void k_blend(const float* __restrict__ covt,
                        const float* __restrict__ scal,
                        float* __restrict__ sig0, float* __restrict__ sig1) {
  const size_t idx = (size_t)blockIdx.x * 256 + threadIdx.x;
  const int i = (int)(idx / DD), j = (int)(idx % DD);
  const float ct = covt[idx] * (1.f / (float)(NSUP - 1));
  const float eye = (i == j) ? 1.f : 0.f;
  {
    const float n = scal[0], lam = n / (n + 1.f);
    sig0[idx] = lam * (sig0[idx] / (n - 1.f)) + (1.f - lam) * ct + eye;
  }
  {
    const float n = scal[1], lam = n / (n + 1.f);
    sig1[idx] = lam * (sig1[idx] / (n - 1.f)) + (1.f - lam) * ct + eye;
  }
}

// ||sigma||_1 = max column abs-sum (atomicMax on positive-float bits)
__global__ void k_colnorm1(const float* __restrict__ sig,
                           unsigned* __restrict__ normbits) {
  const int j = blockIdx.x * 256 + threadIdx.x;
  float s = 0.f;
  for (int i = 0; i < DD; ++i) s += fabsf(sig[(size_t)i * DD + j]);
  atomicMax(normbits, __float_as_uint(s));
}

// Newton-Schulz init: X0 = (2/(1+||A||_1)) * I  (valid since lambda_min >= 1)
__global__ void k_diaginit(const unsigned* __restrict__ normbits,
                           float* __restrict__ X) {
  const size_t idx = (size_t)blockIdx.x * 256 + threadIdx.x;
  const float c0 = 2.f / (1.f + __uint_as_float(*normbits));
  X[idx] = ((idx / DD) == (idx % DD)) ? c0 : 0.f;
}

// Fused logits: logit[q,c] = - (xq-mu)^T P (xq-mu).
// Block = 64 queries; loops all 16 column-chunks of P; deterministic stores.
__global__ __launch_bounds__(256)
void k_logits(const float* __restrict__ G, const int* __restrict__ qidx,
              const float* __restrict__ prec, const float* __restrict__ mu,
              int c, float* __restrict__ out) {
  __shared__ float As[64 * 18];
  __shared__ float Bs[16 * 136];
  __shared__ int qs[64];
  __shared__ float lpart[64][2];
  const int t = threadIdx.x, lane = t & 31, wv = t >> 5;
  const int wm = wv >> 1, wn = wv & 1;     // wm: 16 query rows; wn: 64 cols
  const int hi = lane >> 4, ln = lane & 15;
  const int q0 = blockIdx.x * 64;
  if (t < 64) qs[t] = qidx[q0 + t];
  __syncthreads();
  float part[8];
#pragma unroll
  for (int v = 0; v < 8; ++v) part[v] = 0.f;
  for (int n0 = 0; n0 < DD; n0 += 128) {
    v8f acc[4];
#pragma unroll
    for (int j = 0; j < 4; ++j) acc[j] = {};
    for (int kk = 0; kk < DD; kk += 16) {
      {                                     // A: 64x16 centered query tile
        const int r = t >> 2, kg = (t & 3) << 2;
        const float* gp = &G[(size_t)qs[r] * DD + kk + kg];
        const float* mp = &mu[kk + kg];
        As[r * 18 + kg + 0] = gp[0] - mp[0];
        As[r * 18 + kg + 1] = gp[1] - mp[1];
        As[r * 18 + kg + 2] = gp[2] - mp[2];
        As[r * 18 + kg + 3] = gp[3] - mp[3];
      }
#pragma unroll
      for (int p = 0; p < 8; ++p) {         // B: prec[k][n] tile
        const int k = p * 2 + (t >> 7);
        const int n = t & 127;
        Bs[k * 136 + n] = prec[(size_t)(kk + k) * DD + n0 + n];
      }
      __syncthreads();
#pragma unroll
      for (int k0 = 0; k0 < 16; k0 += 4) {
        const v2f a = *(const v2f*)&As[(wm * 16 + ln) * 18 + k0 + 2 * hi];
#pragma unroll
        for (int nt = 0; nt < 4; ++nt) {
          const int n = wn * 64 + nt * 16 + ln;
          v2f b;
          b.x = Bs[(k0 + 2 * hi) * 136 + n];
          b.y = Bs[(k0 + 2 * hi + 1) * 136 + n];
          acc[nt] = wmma4(a, b, acc[nt]);
        }
      }
      __syncthreads();
    }
    // partial quadratic form over this column chunk
#pragma unroll
    for (int nt = 0; nt < 4; ++nt) {
      const int n = n0 + wn * 64 + nt * 16 + ln;
      const float mun = mu[n];
#pragma unroll
      for (int v = 0; v < 8; ++v) {
        const int ql = wm * 16 + hi * 8 + v;
        const float dq = G[(size_t)qs[ql] * DD + n] - mun;
        part[v] += acc[nt][v] * dq;
      }
    }
  }
#pragma unroll
  for (int v = 0; v < 8; ++v) {             // reduce over 16-lane group
    float s = part[v];
    s += __shfl_xor(s, 1, 32);
    s += __shfl_xor(s, 2, 32);
    s += __shfl_xor(s, 4, 32);
    s += __shfl_xor(s, 8, 32);
    part[v] = s;
  }
  if (ln == 0) {
#pragma unroll
    for (int v = 0; v < 8; ++v) lpart[wm * 16 + hi * 8 + v][wn] = part[v];
  }
  __syncthreads();
  if (t < 64) out[(size_t)(q0 + t) * NCLS + c] = -(lpart[t][0] + lpart[t][1]);
}

// ---------------------------------------------------------------------------
extern "C" void kernel_launch(void* const* d_in, const int* in_sizes, int n_in,
                              void* d_out, int out_size, void* d_ws,
                              size_t ws_size, hipStream_t stream) {
  const float* G = (const float*)d_in[0];
  const int* labels = (const int*)d_in[1];
  const unsigned char* isq = (const unsigned char*)d_in[2];
  float* out = (float*)d_out;
  char* ws = (char*)d_ws;

  int* sidx = (int*)(ws + OFF_SIDX);
  int* qidx = (int*)(ws + OFF_QIDX);
  float* scal = (float*)(ws + OFF_SCAL);
  unsigned* norms = (unsigned*)(ws + OFF_SCAL) + 4;
  float* mu = (float*)(ws + OFF_MU);
  float* meanall = (float*)(ws + OFF_MEAN);
  float* covt = (float*)(ws + OFF_COVT);
  float* sig[2] = {(float*)(ws + OFF_SIG0), (float*)(ws + OFF_SIG1)};
  float* Xa = (float*)(ws + OFF_XA);
  float* Xb = (float*)(ws + OFF_XB);
  float* T = (float*)(ws + OFF_TMP);

  k_init<<<1, 64, 0, stream>>>(norms);
  k_indices<<<1, 256, 0, stream>>>(labels, isq, sidx, qidx, scal);
  k_colstats<<<DD / 256, 256, 0, stream>>>(G, labels, sidx, scal, mu, meanall);

  dim3 g16(DD / 128, DD / 128);
  k_gemm_cov<<<g16, 256, 0, stream>>>(G, sidx, labels, meanall, meanall, -1, covt);
  k_gemm_cov<<<g16, 256, 0, stream>>>(G, sidx, labels, mu, mu, 0, sig[0]);
  k_gemm_cov<<<g16, 256, 0, stream>>>(G, sidx, labels, mu + DD, mu + DD, 1, sig[1]);
  k_blend<<<(DD * DD) / 256, 256, 0, stream>>>(covt, scal, sig[0], sig[1]);

  for (int c = 0; c < NCLS; ++c) {
    k_colnorm1<<<DD / 256, 256, 0, stream>>>(sig[c], norms + c);
    k_diaginit<<<(DD * DD) / 256, 256, 0, stream>>>(norms + c, Xa);
    float* Xc = Xa;
    float* Xn = Xb;
    for (int it = 0; it < NS_ITERS; ++it) {
      k_gemm_sq<<<g16, 256, 0, stream>>>(sig[c], Xc, Xc, 1.f, 0.f, T);
      k_gemm_sq<<<g16, 256, 0, stream>>>(Xc, T, Xc, -1.f, 2.f, Xn);
      float* tmp = Xc; Xc = Xn; Xn = tmp;
    }
    k_logits<<<NQRY / 64, 256, 0, stream>>>(G, qidx, Xc, mu + c * DD, c, out);
  }
}